// ParallelExperts_75428215653130
// MI455X (gfx1250) — compile-verified
//
#include <hip/hip_runtime.h>

typedef float v2f __attribute__((ext_vector_type(2)));
typedef float v8f __attribute__((ext_vector_type(8)));
typedef int   v4i __attribute__((__vector_size__(4 * sizeof(int))));

#define N_TOKENS 8192
#define TOPK     2
#define NK       (N_TOKENS * TOPK)   // 16384 sorted rows
#define D_IN     2048
#define D_OUT    2048

#define BM 128
#define BN 128
#define BK 16
#define LDA 20        // padded LDS row stride: bank-conflict-free, 16B-aligned
#define NTHREADS 256

// --------------------------- CDNA5 async memory->LDS -----------------------
#if defined(__has_builtin)
#  if __has_builtin(__builtin_amdgcn_global_load_async_to_lds_b128)
#    define HAVE_ASYNC_LDS 1
#  endif
#endif
#ifndef HAVE_ASYNC_LDS
#  define HAVE_ASYNC_LDS 0
#endif

#if HAVE_ASYNC_LDS
typedef __attribute__((address_space(1))) v4i* gas_v4i_ptr;
typedef __attribute__((address_space(3))) v4i* las_v4i_ptr;
#endif

__device__ __forceinline__ void async_cp_b128(void* lds_dst, const void* gsrc) {
#if HAVE_ASYNC_LDS
    __builtin_amdgcn_global_load_async_to_lds_b128(
        (gas_v4i_ptr)(void*)gsrc, (las_v4i_ptr)lds_dst, /*offset=*/0, /*cpol=*/0);
#else
    (void)lds_dst; (void)gsrc;
#endif
}

// ---------------------------------------------------------------------------
// Kernel 0: zero the output (harness poisons d_out; atomics need a 0 base).
// ---------------------------------------------------------------------------
__global__ __launch_bounds__(256) void moe_zero_out(float4* __restrict__ p, int n4) {
    int i = blockIdx.x * blockDim.x + threadIdx.x;
    if (i < n4) p[i] = make_float4(0.f, 0.f, 0.f, 0.f);
}

// --------------------------- WMMA stage over one LDS buffer ----------------
__device__ __forceinline__ void mma_stage(const float (*__restrict__ A)[LDA],
                                          const float (*__restrict__ B)[LDA],
                                          v8f (&acc)[2][4],
                                          int wm, int wn, int l_lo, int l_hi) {
    #pragma unroll
    for (int kk = 0; kk < BK; kk += 4) {
        const int kl = kk + 2 * l_hi;       // even -> 8B-aligned ds_read_b64
        v2f afr[2], bfr[4];
        afr[0] = *(const v2f*)&A[wm + l_lo][kl];
        afr[1] = *(const v2f*)&A[wm + 16 + l_lo][kl];
        #pragma unroll
        for (int nt = 0; nt < 4; ++nt)
            bfr[nt] = *(const v2f*)&B[wn + nt * 16 + l_lo][kl];
        #pragma unroll
        for (int mt = 0; mt < 2; ++mt)
            #pragma unroll
            for (int nt = 0; nt < 4; ++nt)
                acc[mt][nt] = __builtin_amdgcn_wmma_f32_16x16x4_f32(
                    false, afr[mt], false, bfr[nt],
                    (short)0, acc[mt][nt], false, false);
    }
}

// ----------------- gated atomic combine (exactly 2 adds per out elem) ------
// C/D layout: lane l holds col n=l&15, rows m = v + 8*(l>>4), v=0..7
__device__ __forceinline__ void epilogue(const v8f (&acc)[2][4], int e,
                                         const int* __restrict__ e_s,
                                         const int* __restrict__ tok_s,
                                         const float* __restrict__ gate_s,
                                         float* __restrict__ out,
                                         int n0, int wm, int wn, int l_lo, int l_hi) {
    #pragma unroll
    for (int mt = 0; mt < 2; ++mt) {
        #pragma unroll
        for (int v = 0; v < 8; ++v) {
            const int ml = wm + mt * 16 + v + 8 * l_hi;
            if (e_s[ml] == e) {
                const float g = gate_s[ml];
                float* orow = out + (size_t)tok_s[ml] * D_OUT + n0;
                #pragma unroll
                for (int nt = 0; nt < 4; ++nt)
                    unsafeAtomicAdd(orow + wn + nt * 16 + l_lo, g * acc[mt][nt][v]);
            }
        }
    }
}

// ---------------------------------------------------------------------------
// Kernel 1: grouped scatter GEMM over expert-sorted rows + gated combine.
//   sorted row p: flat j = ssi[p], token = j>>1, expert = sei[p], gate = gates[j]
//   out[token, :] += gate * x[token, :] @ W[e, :, :]^T      (atomic, 2 adds/elem)
// ---------------------------------------------------------------------------
__global__ __launch_bounds__(NTHREADS) void moe_scatter_gemm(
    const float* __restrict__ x,      // [N_TOKENS, D_IN]
    const float* __restrict__ w,      // [E, D_OUT, D_IN]
    const float* __restrict__ gates,  // [N_TOKENS, TOPK] flat == flat j
    const int*   __restrict__ sei,    // [NK] sorted expert ids
    const int*   __restrict__ ssi,    // [NK] sorted scattered idxs
    float*       __restrict__ out)    // [N_TOKENS, D_OUT]
{
    __shared__ __align__(16) float As[2][BM][LDA];   // double-buffered A
    __shared__ __align__(16) float Bs[2][BN][LDA];   // double-buffered B
    __shared__ int   tok_s[BM];
    __shared__ float gate_s[BM];
    __shared__ int   e_s[BM];

    const int tid = threadIdx.x;
    const int m0  = blockIdx.y * BM;     // sorted-row base
    const int n0  = blockIdx.x * BN;     // output-column base

    if (tid < BM) {
        const int p = m0 + tid;
        const int j = ssi[p];
        tok_s[tid]  = j >> 1;            // TOPK == 2
        gate_s[tid] = gates[j];          // gates flat index equals j
        e_s[tid]    = sei[p];
    }
    __syncthreads();

    const int e_lo = e_s[0];
    const int e_hi = e_s[BM - 1];        // sorted => contiguous expert range

    const int lane = tid & 31;
    const int wid  = tid >> 5;           // wave 0..7
    const int l_lo = lane & 15;
    const int l_hi = lane >> 4;
    const int wm   = (wid & 3) * 32;     // 4 waves over BM=128
    const int wn   = (wid >> 2) * 64;    // 2 waves over BN=128

    // Cooperative load assignment (A and B identical: 128 rows x 16 floats).
    const int ar = tid >> 1;             // row 0..127
    const int ah = (tid & 1) * 8;        // 8-float half of the BK=16 slab

    const float* arow = x + (size_t)tok_s[ar] * D_IN + ah;

#if HAVE_ASYNC_LDS
    if (e_lo == e_hi) {
        // ------------- hot path: single-expert tile, async double-buffered --
        const int e = e_lo;
        const float* brow = w + (size_t)e * D_OUT * D_IN
                              + (size_t)(n0 + ar) * D_IN + ah;
        float* const asd[2] = { &As[0][ar][ah], &As[1][ar][ah] };
        float* const bsd[2] = { &Bs[0][ar][ah], &Bs[1][ar][ah] };

        v8f acc[2][4];
        #pragma unroll
        for (int i = 0; i < 2; ++i)
            #pragma unroll
            for (int q = 0; q < 4; ++q)
                #pragma unroll
                for (int v = 0; v < 8; ++v) acc[i][q][v] = 0.0f;

        // prologue: stage 0 into buffer 0 (4 asyncs per lane: 2xA + 2xB)
        async_cp_b128(asd[0],     arow);
        async_cp_b128(asd[0] + 4, arow + 4);
        async_cp_b128(bsd[0],     brow);
        async_cp_b128(bsd[0] + 4, brow + 4);

        for (int k0 = 0; k0 < D_IN; k0 += BK) {
            const int pb = (k0 >> 4) & 1;
            const int qb = pb ^ 1;
            __syncthreads();                       // buf qb free of readers
            if (k0 + BK < D_IN) {
                const int k1 = k0 + BK;
                async_cp_b128(asd[qb],     arow + k1);
                async_cp_b128(asd[qb] + 4, arow + k1 + 4);
                async_cp_b128(bsd[qb],     brow + k1);
                async_cp_b128(bsd[qb] + 4, brow + k1 + 4);
                // oldest group (stage k0, 4 asyncs) done; next 4 may remain
                asm volatile("s_wait_asynccnt 4" ::: "memory");
            } else {
                asm volatile("s_wait_asynccnt 0" ::: "memory");
            }
            __syncthreads();                       // stage k0 visible to all
            mma_stage(As[pb], Bs[pb], acc, wm, wn, l_lo, l_hi);
        }
        epilogue(acc, e, e_s, tok_s, gate_s, out, n0, wm, wn, l_lo, l_hi);
        return;
    }
#endif

    // ------------- boundary tiles (and no-async fallback): masked reg path --
    for (int e = e_lo; e <= e_hi; ++e) {
        v8f acc[2][4];
        #pragma unroll
        for (int i = 0; i < 2; ++i)
            #pragma unroll
            for (int q = 0; q < 4; ++q)
                #pragma unroll
                for (int v = 0; v < 8; ++v) acc[i][q][v] = 0.0f;

        const bool   a_valid = (e_s[ar] == e);
        const float* brow = w + (size_t)e * D_OUT * D_IN
                              + (size_t)(n0 + ar) * D_IN + ah;
        float* const as_dst = &As[0][ar][ah];
        float* const bs_dst = &Bs[0][ar][ah];

        for (int k0 = 0; k0 < D_IN; k0 += BK) {
            __syncthreads();
            float4 av0, av1;
            if (a_valid) {
                av0 = *(const float4*)(arow + k0);
                av1 = *(const float4*)(arow + k0 + 4);
            } else {
                av0 = make_float4(0.f, 0.f, 0.f, 0.f);
                av1 = av0;
            }
            *(float4*)as_dst       = av0;
            *(float4*)(as_dst + 4) = av1;
            *(float4*)bs_dst       = *(const float4*)(brow + k0);
            *(float4*)(bs_dst + 4) = *(const float4*)(brow + k0 + 4);
            if (k0 + BK < D_IN) {
                __builtin_prefetch(arow + k0 + BK, 0, 3);
                __builtin_prefetch(brow + k0 + BK, 0, 3);
            }
            __syncthreads();
            mma_stage(As[0], Bs[0], acc, wm, wn, l_lo, l_hi);
        }
        epilogue(acc, e, e_s, tok_s, gate_s, out, n0, wm, wn, l_lo, l_hi);
    }
}

// ---------------------------------------------------------------------------
extern "C" void kernel_launch(void* const* d_in, const int* in_sizes, int n_in,
                              void* d_out, int out_size, void* d_ws, size_t ws_size,
                              hipStream_t stream) {
    const float* x     = (const float*)d_in[0];   // inputs [8192, 2048] f32
    const float* w     = (const float*)d_in[1];   // expert_weights [8, 2048, 2048] f32
    const float* gates = (const float*)d_in[2];   // gates [8192, 2] f32
    // d_in[3] = k (==2, hardcoded)
    const int* sei = (const int*)d_in[4];         // sorted_expert_idxs [16384]
    const int* ssi = (const int*)d_in[5];         // sorted_scattered_idxs [16384]
    // d_in[6] = expert_offsets (unused; boundaries derived from sei per tile)
    float* out = (float*)d_out;

    const int n4 = (N_TOKENS * D_OUT) / 4;
    moe_zero_out<<<dim3((n4 + 255) / 256), dim3(256), 0, stream>>>((float4*)out, n4);

    dim3 grid(D_OUT / BN, NK / BM);               // (16, 128)
    moe_scatter_gemm<<<grid, dim3(NTHREADS), 0, stream>>>(x, w, gates, sei, ssi, out);
}